// CustomLSTM_8873402434090
// MI455X (gfx1250) — compile-verified
//
#include <hip/hip_runtime.h>
#include <hip/hip_bf16.h>

// LSTM forward on gfx1250 (MI455X): per-batch-row recurrence partitioned over
// 16 workgroups; per-step 16x64 @ 64x256 GEMM done with V_WMMA_F32_16X16X4_F32.
// Activations use v_exp_f32 + v_rcp_f32 (no IEEE divide on the critical path).

#define SEQ 2048
#define BB  256
#define INW 2
#define HH  64
#define G4  256   // 4*H

#define HPAD 68   // row pitch (dwords) for LDS matrices; avoids bank conflicts

typedef __attribute__((ext_vector_type(2))) float v2f;
typedef __attribute__((ext_vector_type(8))) float v8f;

__device__ __forceinline__ float fast_rcp(float x) {
    return __builtin_amdgcn_rcpf(x);          // v_rcp_f32
}
__device__ __forceinline__ float fast_sigmoid(float x) {
    return fast_rcp(1.0f + __expf(-x));       // v_exp_f32 + v_rcp_f32
}
__device__ __forceinline__ float fast_tanh(float x) {
    // tanh(x) = 2*sigmoid(2x) - 1; saturates correctly at +/-1
    return fmaf(2.0f, fast_rcp(1.0f + __expf(-2.0f * x)), -1.0f);
}

__global__ __launch_bounds__(512)
void lstm_wmma_f32_kernel(const float* __restrict__ input,   // [SEQ][B][IN]
                          const float* __restrict__ W_ih,    // [IN][4H]
                          const float* __restrict__ W_hh,    // [H][4H]
                          const float* __restrict__ b_ih,    // [4H]
                          const float* __restrict__ b_hh,    // [4H]
                          float* __restrict__ out)           // [SEQ*B*H | B*H | B*H]
{
    __shared__ float h_buf[16 * HPAD];      // h_prev block, row-major [16][HPAD]
    __shared__ float act[4][16 * HPAD];     // activated i,f,g,o tiles [16][64] padded
    __shared__ float xin[32];               // input[s, bbase..bbase+15, 0..1]

    const int tid   = threadIdx.x;
    const int wave  = tid >> 5;             // 0..15 : owns gate-column tile wave*16..+15
    const int lane  = tid & 31;
    const int lhalf = lane >> 4;            // 0/1
    const int lmod  = lane & 15;
    const int bbase = blockIdx.x * 16;      // this block's batch rows

    // ---- preload loop-invariant W_hh B-fragments: 16 K-tiles x 2 VGPRs ----
    // B frag layout (4x16 f32): lane L, vgpr v -> B[K = v + 2*(L/16)][N = L%16]
    float bw0[16], bw1[16];
    {
        const int n = wave * 16 + lmod;
        #pragma unroll
        for (int t = 0; t < 16; ++t) {
            const int k = t * 4 + 2 * lhalf;
            bw0[t] = W_hh[(size_t)k       * G4 + n];
            bw1[t] = W_hh[(size_t)(k + 1) * G4 + n];
        }
    }
    // per-lane input-projection constants for this lane's gate column
    const int   gcol = wave * 16 + lmod;
    const float wi0  = W_ih[gcol];
    const float wi1  = W_ih[G4 + gcol];
    const float bias = b_ih[gcol] + b_hh[gcol];

    // wave-uniform, loop-invariant activation constants:
    //   sigmoid waves:  v = 1*sigmoid(1*x) + 0
    //   tanh waves (g): v = 2*sigmoid(2*x) - 1
    const int   gate    = wave >> 2;        // 0:i 1:f 2:g 3:o
    const float a_scale = (gate == 2) ? -2.0f : -1.0f;   // folded negation for exp
    const float a_mul   = (gate == 2) ?  2.0f :  1.0f;
    const float a_add   = (gate == 2) ? -1.0f :  0.0f;

    // zero h0 block, stage input for s=0
    for (int i = tid; i < 16 * HPAD; i += 512) h_buf[i] = 0.0f;
    if (tid < 32) xin[tid] = input[(size_t)bbase * INW + tid];

    // c-state lives in the update threads' registers:
    //   thread t owns rows um = t>>5, cols uj = (t&31)*2 .. +1
    const int um = tid >> 5;
    const int uj = (tid & 31) * 2;
    float c0 = 0.0f, c1 = 0.0f;

    float* out_h = out + (size_t)SEQ * BB * HH;
    float* out_c = out_h + (size_t)BB * HH;

    for (int s = 0; s < SEQ; ++s) {
        __syncthreads();   // h_buf + xin for step s are ready

        // ---- gates tile: acc = x @ W_ih + b  ;  acc += h_prev @ W_hh ----
        v8f acc, acc2;
        #pragma unroll
        for (int r = 0; r < 8; ++r) {
            const int bl = r + 8 * lhalf;              // batch row of acc[r]
            const float x0 = xin[bl * 2];
            const float x1 = xin[bl * 2 + 1];
            acc[r]  = fmaf(x0, wi0, fmaf(x1, wi1, bias));
            acc2[r] = 0.0f;
        }
        // A frag layout (16x4 f32): lane L, vgpr v -> A[M = L%16][K = v + 2*(L/16)]
        #pragma unroll
        for (int t = 0; t < 16; t += 2) {
            {
                const int k0 = t * 4 + 2 * lhalf;
                v2f a = *(const v2f*)&h_buf[lmod * HPAD + k0];
                v2f b = { bw0[t], bw1[t] };
                acc = __builtin_amdgcn_wmma_f32_16x16x4_f32(
                          false, a, false, b, (short)0, acc, false, false);
            }
            {
                const int k0 = (t + 1) * 4 + 2 * lhalf;
                v2f a = *(const v2f*)&h_buf[lmod * HPAD + k0];
                v2f b = { bw0[t + 1], bw1[t + 1] };
                acc2 = __builtin_amdgcn_wmma_f32_16x16x4_f32(
                          false, a, false, b, (short)0, acc2, false, false);
            }
        }

        // ---- activations (branchless, wave-uniform), staged to LDS ----
        {
            float* dst = &act[gate][0];
            const int j = (wave & 3) * 16 + lmod;      // column within H
            #pragma unroll
            for (int r = 0; r < 8; ++r) {
                const int m = r + 8 * lhalf;
                const float v = acc[r] + acc2[r];
                const float t = fast_rcp(1.0f + __expf(a_scale * v));
                dst[m * HPAD + j] = fmaf(a_mul, t, a_add);
            }
        }

        __syncthreads();   // act ready; all A-fragment reads of h_buf done

        // ---- elementwise cell update; write h to LDS + global ----
        {
            const float2 iv = *(const float2*)&act[0][um * HPAD + uj];
            const float2 fv = *(const float2*)&act[1][um * HPAD + uj];
            const float2 gv = *(const float2*)&act[2][um * HPAD + uj];
            const float2 ov = *(const float2*)&act[3][um * HPAD + uj];
            c0 = fmaf(fv.x, c0, iv.x * gv.x);
            c1 = fmaf(fv.y, c1, iv.y * gv.y);
            const float h0 = ov.x * fast_tanh(c0);
            const float h1 = ov.y * fast_tanh(c1);
            h_buf[um * HPAD + uj]     = h0;
            h_buf[um * HPAD + uj + 1] = h1;
            const float2 hv = make_float2(h0, h1);
            *(float2*)&out[(size_t)s * BB * HH + (size_t)(bbase + um) * HH + uj] = hv;

            // stage next step's input; prefetch a few steps ahead
            if (tid < 32) {
                if (s + 1 < SEQ)
                    xin[tid] = input[(size_t)(s + 1) * BB * INW + (size_t)bbase * INW + tid];
                if (s + 8 < SEQ)
                    __builtin_prefetch(&input[(size_t)(s + 8) * BB * INW + (size_t)bbase * INW + tid], 0, 0);
            }
            if (s == SEQ - 1) {
                *(float2*)&out_h[(size_t)(bbase + um) * HH + uj] = hv;
                *(float2*)&out_c[(size_t)(bbase + um) * HH + uj] = make_float2(c0, c1);
            }
        }
    }
}

extern "C" void kernel_launch(void* const* d_in, const int* in_sizes, int n_in,
                              void* d_out, int out_size, void* d_ws, size_t ws_size,
                              hipStream_t stream) {
    const float* input = (const float*)d_in[0];
    const float* W_ih  = (const float*)d_in[1];
    const float* W_hh  = (const float*)d_in[2];
    const float* b_ih  = (const float*)d_in[3];
    const float* b_hh  = (const float*)d_in[4];
    float* out = (float*)d_out;

    lstm_wmma_f32_kernel<<<dim3(BB / 16), dim3(512), 0, stream>>>(
        input, W_ih, W_hh, b_ih, b_hh, out);
}